// Universal_Decoder_37134287241755
// MI455X (gfx1250) — compile-verified
//
#include <hip/hip_runtime.h>
#include <hip/hip_bf16.h>
#include <math.h>
#include <stdint.h>

// ---- problem constants (match reference) ----
#define LL    2
#define HD    512
#define ZD    512
#define VD    50257
#define TT    128
#define ZH    1024      // Z + H
#define G3H   1536      // 3*H
#define BPAD  516       // LDS B-tile row stride (floats): 516 mod 64 = 4 -> conflict-free

typedef float v2f __attribute__((ext_vector_type(2)));
typedef float v8f __attribute__((ext_vector_type(8)));

// ---------------------------------------------------------------------------
// Kernel 1: recover one-hot index per timestep.
//   inp at step t = sos (t==0) else target[t-1].  Exactly one element is 1.0.
// ---------------------------------------------------------------------------
__global__ void find_idx_kernel(const float* __restrict__ target,  // [T,V]
                                const float* __restrict__ sos,     // [V]
                                int* __restrict__ idx) {           // [T]
    int t = blockIdx.x;
    const float* row = (t == 0) ? sos : (target + (size_t)(t - 1) * VD);
    for (int i = threadIdx.x; i < VD; i += blockDim.x) {
        if (row[i] > 0.5f) idx[t] = i;   // unique hit, no race
    }
}

// ---------------------------------------------------------------------------
// Kernel 2: build layer-0 input X0 [T*2, 1024]:
//   X0[t*2+j][k] = input_variable[j][k]                      (k < 512)
//                = relu(emb_W[j*512 + k-512, idx[t]] + emb_b) (k >= 512)
// ---------------------------------------------------------------------------
__global__ void build_x_kernel(const float* __restrict__ inputv,  // [2,512]
                               const float* __restrict__ embW,    // [1024,V]
                               const float* __restrict__ embB,    // [1024]
                               const int*   __restrict__ idx,     // [T]
                               float* __restrict__ X0) {          // [256,1024]
    int gid = blockIdx.x * blockDim.x + threadIdx.x;   // 0..256*1024-1
    int row = gid >> 10;
    int k   = gid & 1023;
    int t = row >> 1, j = row & 1;
    float v;
    if (k < ZD) {
        v = inputv[j * ZD + k];
    } else {
        int e = j * HD + (k - ZD);
        float x = embW[(size_t)e * VD + idx[t]] + embB[e];
        v = x > 0.0f ? x : 0.0f;
    }
    X0[gid] = v;
}

// ---------------------------------------------------------------------------
// Kernel 3: GI0[256,1536] = X0[256,1024] @ W_ih0^T + b_ih0, fp32 WMMA 16x16x4.
//   One wave per 16x16 output tile; 16 M-tiles x 96 N-tiles = 1536 waves.
//   A frag (16x4, MxK): lane%16 = M, VGPRv + half -> K = v + 2*half.
//   B frag (4x16, KxN): lane%16 = N, VGPRv + half -> K = v + 2*half.
//   C frag (16x16):     lane%16 = N, VGPRv -> M = v + 8*half.
// ---------------------------------------------------------------------------
__global__ void gemm_gi0_kernel(const float* __restrict__ X0,   // [256,1024]
                                const float* __restrict__ W,    // [1536,1024]
                                const float* __restrict__ b,    // [1536]
                                float* __restrict__ GI0) {      // [256,1536]
    int wave = (blockIdx.x * blockDim.x + threadIdx.x) >> 5;    // 0..1535
    int lane = threadIdx.x & 31;
    int half = lane >> 4;
    int l16  = lane & 15;
    int mBase = (wave & 15) * 16;
    int nBase = (wave >> 4) * 16;

    const float* arow = X0 + (size_t)(mBase + l16) * ZH;
    const float* brow = W  + (size_t)(nBase + l16) * ZH;
    __builtin_prefetch(brow, 0, 1);   // global_prefetch of the weight row

    v8f c = {};
    for (int k0 = 0; k0 < ZH; k0 += 4) {
        v2f a, bf;
        a.x  = arow[k0 + half * 2 + 0];
        a.y  = arow[k0 + half * 2 + 1];
        bf.x = brow[k0 + half * 2 + 0];
        bf.y = brow[k0 + half * 2 + 1];
        c = __builtin_amdgcn_wmma_f32_16x16x4_f32(false, a, false, bf,
                                                  (short)0, c, false, false);
    }
    float bias = b[nBase + l16];
#pragma unroll
    for (int v = 0; v < 8; ++v) {
        int m = mBase + half * 8 + v;
        GI0[(size_t)m * G3H + nBase + l16] = c[v] + bias;
    }
}

// ---------------------------------------------------------------------------
// Kernel 4: sequential GRU chain (the only truly serial part).
//   Single workgroup, 1024 threads (32 waves). State in LDS.
//   float4 weight loads (global_load_b128) + float4 LDS broadcasts.
// ---------------------------------------------------------------------------
__device__ __forceinline__ float dot512(const float* __restrict__ wrow,
                                        const float* __restrict__ vec,
                                        float acc) {
    const float4* w4 = reinterpret_cast<const float4*>(wrow);
    const float4* v4 = reinterpret_cast<const float4*>(vec);
#pragma unroll 4
    for (int k = 0; k < HD / 4; ++k) {
        float4 w = w4[k];
        float4 h = v4[k];
        acc += w.x * h.x + w.y * h.y + w.z * h.z + w.w * h.w;
    }
    return acc;
}

__global__ __launch_bounds__(1024)
void gru_seq_kernel(const float* __restrict__ GI0,    // [256,1536] incl b_ih0
                    const float* __restrict__ Whh0,   // [1536,512]
                    const float* __restrict__ bhh0,   // [1536]
                    const float* __restrict__ Wih1,   // [1536,512]
                    const float* __restrict__ bih1,   // [1536]
                    const float* __restrict__ Whh1,   // [1536,512]
                    const float* __restrict__ bhh1,   // [1536]
                    float* __restrict__ Hout) {       // [128,512]
    __shared__ __align__(16) float h0[HD], h1[HD], o00[HD], o01[HD];
    __shared__ float gh[G3H], gi1[G3H], gh1[G3H];

    const int tid = threadIdx.x;
    if (tid < HD) { h0[tid] = 0.0f; h1[tid] = 0.0f; }
    __syncthreads();

    for (int t = 0; t < TT; ++t) {
        // ----- layer 0, inner times j = 0,1 -----
        for (int j = 0; j < 2; ++j) {
            const float* gi = GI0 + (size_t)(t * 2 + j) * G3H;
            gh[tid] = dot512(Whh0 + (size_t)tid * HD, h0, bhh0[tid]);
            if (tid < HD) {
                int r2 = tid + 1024;
                gh[r2] = dot512(Whh0 + (size_t)r2 * HD, h0, bhh0[r2]);
            }
            __syncthreads();
            if (tid < HD) {
                int k = tid;
                float r  = 1.0f / (1.0f + expf(-(gi[k]        + gh[k])));
                float z  = 1.0f / (1.0f + expf(-(gi[HD + k]   + gh[HD + k])));
                float nn = tanhf(gi[2 * HD + k] + r * gh[2 * HD + k]);
                float hn = (1.0f - z) * nn + z * h0[k];
                h0[k] = hn;
                if (j == 0) o00[k] = hn; else o01[k] = hn;
            }
            __syncthreads();
        }
        // ----- layer 1, inner times j = 0,1 -----
        for (int j = 0; j < 2; ++j) {
            const float* x = (j == 0) ? o00 : o01;
            gi1[tid] = dot512(Wih1 + (size_t)tid * HD, x,  bih1[tid]);
            gh1[tid] = dot512(Whh1 + (size_t)tid * HD, h1, bhh1[tid]);
            if (tid < HD) {
                int r2 = tid + 1024;
                gi1[r2] = dot512(Wih1 + (size_t)r2 * HD, x,  bih1[r2]);
                gh1[r2] = dot512(Whh1 + (size_t)r2 * HD, h1, bhh1[r2]);
            }
            __syncthreads();
            if (tid < HD) {
                int k = tid;
                float r  = 1.0f / (1.0f + expf(-(gi1[k]        + gh1[k])));
                float z  = 1.0f / (1.0f + expf(-(gi1[HD + k]   + gh1[HD + k])));
                float nn = tanhf(gi1[2 * HD + k] + r * gh1[2 * HD + k]);
                float hn = (1.0f - z) * nn + z * h1[k];
                h1[k] = hn;
                if (j == 0) Hout[(size_t)t * HD + k] = hn;   // probs source
            }
            __syncthreads();
        }
    }
}

// ---------------------------------------------------------------------------
// Kernel 5: logits[128,V] = Hout[128,512] @ out_W^T + out_b, fp32 WMMA.
//   Block = 8 waves = 8 M-tiles x 1 N-tile; grid = ceil(V/16) = 3142 blocks.
//   The 16x512 out_W panel is staged in LDS ONCE per block using CDNA5
//   async copy (global_load_async_to_lds_b128, ASYNCcnt), removing the 8x
//   redundant L2 reads.  LDS row stride padded (516) -> bank-conflict-free.
// ---------------------------------------------------------------------------
__global__ __launch_bounds__(256)
void gemm_out_kernel(const float* __restrict__ Hout,  // [128,512]
                     const float* __restrict__ W,     // [V,512]
                     const float* __restrict__ b,     // [V]
                     float* __restrict__ out) {       // [T*V] logits
    __shared__ __align__(16) float tileB[16 * BPAD];  // ~33 KB

    int lane = threadIdx.x & 31;
    int half = lane >> 4;
    int l16  = lane & 15;
    int mBase = (threadIdx.x >> 5) * 16;      // wave -> M tile (0..7)
    int nBase = blockIdx.x * 16;              // N tile
    int n     = nBase + l16;

    // ---- stage B panel: 16 rows x 512 floats = 2048 float4 chunks ----
    for (int c = threadIdx.x; c < 16 * 128; c += 256) {
        int nr = c >> 7;               // 0..15 row within tile
        int kc = (c & 127) * 4;        // K offset
        int ng = nBase + nr;
        float* dst = &tileB[nr * BPAD + kc];
        if (ng < VD) {
            const float* src = W + (size_t)ng * HD + kc;
            uint32_t ldsoff = (uint32_t)(uintptr_t)dst;   // LDS aperture: addr[31:0]
            asm volatile("global_load_async_to_lds_b128 %0, %1, off"
                         :: "v"(ldsoff), "v"(src) : "memory");
        } else {
            dst[0] = 0.0f; dst[1] = 0.0f; dst[2] = 0.0f; dst[3] = 0.0f;
        }
    }
    asm volatile("s_wait_asynccnt 0x0" ::: "memory");
    __syncthreads();

    const float* arow = Hout + (size_t)(mBase + l16) * HD;
    const float* brow = &tileB[l16 * BPAD];

    v8f c = {};
#pragma unroll 8
    for (int k0 = 0; k0 < HD; k0 += 4) {
        v2f a, bf;
        a.x  = arow[k0 + half * 2 + 0];
        a.y  = arow[k0 + half * 2 + 1];
        bf.x = brow[k0 + half * 2 + 0];
        bf.y = brow[k0 + half * 2 + 1];
        c = __builtin_amdgcn_wmma_f32_16x16x4_f32(false, a, false, bf,
                                                  (short)0, c, false, false);
    }
    if (n < VD) {
        float bias = b[n];
#pragma unroll
        for (int v = 0; v < 8; ++v) {
            int m = mBase + half * 8 + v;
            out[(size_t)m * VD + n] = c[v] + bias;
        }
    }
}

// ---------------------------------------------------------------------------
// Kernel 6: per-row softmax (in place over logits in d_out) + argmax -> idxs.
//   jnp.argmax tie-break: lowest index.
// ---------------------------------------------------------------------------
__global__ void softmax_argmax_kernel(float* __restrict__ out) { // [T*V + T]
    int t = blockIdx.x;
    float* row = out + (size_t)t * VD;
    __shared__ float smax[256];
    __shared__ int   sidx[256];
    __shared__ float ssum[256];

    float mx = -INFINITY; int mi = VD;
    for (int i = threadIdx.x; i < VD; i += 256) {
        float v = row[i];
        if (v > mx || (v == mx && i < mi)) { mx = v; mi = i; }
    }
    smax[threadIdx.x] = mx; sidx[threadIdx.x] = mi;
    __syncthreads();
    for (int s = 128; s > 0; s >>= 1) {
        if (threadIdx.x < s) {
            float a = smax[threadIdx.x],  bb = smax[threadIdx.x + s];
            int   ia = sidx[threadIdx.x], ib = sidx[threadIdx.x + s];
            if (bb > a || (bb == a && ib < ia)) {
                smax[threadIdx.x] = bb; sidx[threadIdx.x] = ib;
            }
        }
        __syncthreads();
    }
    float rowmax = smax[0];
    int   rowarg = sidx[0];

    float sum = 0.0f;
    for (int i = threadIdx.x; i < VD; i += 256) sum += expf(row[i] - rowmax);
    ssum[threadIdx.x] = sum;
    __syncthreads();
    for (int s = 128; s > 0; s >>= 1) {
        if (threadIdx.x < s) ssum[threadIdx.x] += ssum[threadIdx.x + s];
        __syncthreads();
    }
    float inv = 1.0f / ssum[0];
    for (int i = threadIdx.x; i < VD; i += 256)
        row[i] = expf(row[i] - rowmax) * inv;

    if (threadIdx.x == 0) out[(size_t)TT * VD + t] = (float)rowarg;
}

// ---------------------------------------------------------------------------
// Host side
// ---------------------------------------------------------------------------
extern "C" void kernel_launch(void* const* d_in, const int* in_sizes, int n_in,
                              void* d_out, int out_size, void* d_ws, size_t ws_size,
                              hipStream_t stream) {
    const float* inputv = (const float*)d_in[0];   // [2,1,512]
    const float* target = (const float*)d_in[1];   // [128,1,V]
    const float* sos    = (const float*)d_in[2];   // [1,V]
    const float* embW   = (const float*)d_in[3];   // [1024,V]
    const float* embB   = (const float*)d_in[4];   // [1024]
    const float* Wih0   = (const float*)d_in[5];   // [1536,1024]
    const float* Whh0   = (const float*)d_in[6];   // [1536,512]
    const float* bih0   = (const float*)d_in[7];   // [1536]
    const float* bhh0   = (const float*)d_in[8];   // [1536]
    const float* Wih1   = (const float*)d_in[9];   // [1536,512]
    const float* Whh1   = (const float*)d_in[10];  // [1536,512]
    const float* bih1   = (const float*)d_in[11];  // [1536]
    const float* bhh1   = (const float*)d_in[12];  // [1536]
    const float* outW   = (const float*)d_in[13];  // [V,512]
    const float* outB   = (const float*)d_in[14];  // [V]
    float* out = (float*)d_out;                    // [T*V] probs + [T] idxs

    // workspace carve-up
    char* ws = (char*)d_ws;
    int*   idx  = (int*)ws;                               //  512 B
    float* X0   = (float*)(ws + 1024);                    //  1 MB   [256,1024]
    float* GI0  = (float*)(ws + 1024 + (size_t)256*ZH*4); //  1.5 MB [256,1536]
    float* Hout = (float*)(ws + 1024 + (size_t)256*ZH*4
                                + (size_t)256*G3H*4);     //  256 KB [128,512]

    find_idx_kernel<<<TT, 256, 0, stream>>>(target, sos, idx);
    build_x_kernel<<<(256 * ZH) / 256, 256, 0, stream>>>(inputv, embW, embB, idx, X0);
    gemm_gi0_kernel<<<192, 256, 0, stream>>>(X0, Wih0, bih0, GI0);
    gru_seq_kernel<<<1, 1024, 0, stream>>>(GI0, Whh0, bhh0, Wih1, bih1, Whh1, bhh1, Hout);
    gemm_out_kernel<<<(VD + 15) / 16, 256, 0, stream>>>(Hout, outW, outB, out);
    softmax_argmax_kernel<<<TT, 256, 0, stream>>>(out);
}